// CDLoss_90348932038721
// MI455X (gfx1250) — compile-verified
//
#include <hip/hip_runtime.h>
#include <math.h>

typedef __attribute__((ext_vector_type(2))) float v2f;
typedef __attribute__((ext_vector_type(8))) float v8f;

// Computes rowmin[b][n] = min_m ||X[b,n,:] - Y[b,m,:]||^2 (clamped >= 0)
// One wave per 16-row tile of X; streams Y in 16-point tiles through
// V_WMMA_F32_16X16X4_F32 with the full distance algebra folded into the
// matrix op:
//   A row = (-2x0, -2x1, -2x2, |x|^2), B col = (y0, y1, y2, 1),
//   C     = |y_col|^2 (splat per lane)
//   => D = x^2 - 2 x.y + y^2  directly from the matrix pipe.
// The M loop is chunked 4 tiles at a time with loads grouped up front so
// they issue as a clause and the loadcnt waits stage down while WMMAs
// overlap. Prefetch is unguarded (speculative prefetch is fault-free).
// Clamp to >= 0 is hoisted out of the loop (max(.,0) commutes with min).
__global__ __launch_bounds__(256) void chamfer_rowmin_wmma(
    const float* __restrict__ X, const float* __restrict__ Y,
    float* __restrict__ rowmin, int Nrows, int Mcols)
{
    const int gtid = blockIdx.x * blockDim.x + threadIdx.x;
    const int wave = gtid >> 5;    // wave32
    const int lane = gtid & 31;
    const int half = lane >> 4;    // selects K pair {0,1} vs {2,3}
    const int rc   = lane & 15;    // A: row index, B/C: column index

    const int tilesPerBatch = Nrows >> 4;
    const int b        = wave / tilesPerBatch;
    const int row_base = (wave % tilesPerBatch) << 4;

    // ---- A operand (persistent across the whole M loop) ----
    const float* xp = X + ((size_t)b * Nrows + row_base + rc) * 3;
    const float x0 = xp[0], x1 = xp[1], x2 = xp[2];
    const float xs = fmaf(x0, x0, fmaf(x1, x1, x2 * x2));
    v2f a;
    a.x = half ? (-2.0f * x2) : (-2.0f * x0);   // K=2 | K=0
    a.y = half ? xs           : (-2.0f * x1);   // K=3 | K=1

    const float INF = __builtin_inff();
    v8f mins = { INF, INF, INF, INF, INF, INF, INF, INF };

    const float* yp = Y + ((size_t)b * Mcols + rc) * 3;

    // 4 tiles (64 points) per chunk; Mcols is a multiple of 64 here.
    for (int m0 = 0; m0 < Mcols; m0 += 64) {
        // ---- group the 4 tile loads so they issue back-to-back ----
        float yv[4][3];
        #pragma unroll
        for (int j = 0; j < 4; ++j) {
            const float* p = yp + (size_t)(m0 + 16 * j) * 3;
            yv[j][0] = p[0];
            yv[j][1] = p[1];
            yv[j][2] = p[2];
        }
        // stream 4 chunks ahead; speculative prefetch never faults
        __builtin_prefetch(yp + (size_t)(m0 + 256) * 3, 0, 3);

        // ---- 4 WMMAs overlap the tail of the load latency ----
        #pragma unroll
        for (int j = 0; j < 4; ++j) {
            const float y0 = yv[j][0], y1 = yv[j][1], y2 = yv[j][2];
            const float ys = fmaf(y0, y0, fmaf(y1, y1, y2 * y2));
            v2f bv;
            bv.x = half ? y2   : y0;            // K=2 | K=0
            bv.y = half ? 1.0f : y1;            // K=3 | K=1

            // C operand carries |y|^2: matrix pipe emits the full distance
            v8f c = { ys, ys, ys, ys, ys, ys, ys, ys };
            c = __builtin_amdgcn_wmma_f32_16x16x4_f32(
                /*neg_a=*/false, a, /*neg_b=*/false, bv,
                /*c_mod=*/(short)0, c, /*reuse_a=*/false, /*reuse_b=*/false);

            #pragma unroll
            for (int i = 0; i < 8; ++i)
                mins[i] = fminf((float)mins[i], (float)c[i]);
        }
    }

    // numerical floor, once (matches reference's maximum(d, 0))
    #pragma unroll
    for (int i = 0; i < 8; ++i)
        mins[i] = fmaxf((float)mins[i], 0.0f);

    // min across the 16 columns: butterfly within each 16-lane half
    #pragma unroll
    for (int m = 1; m <= 8; m <<= 1) {
        #pragma unroll
        for (int i = 0; i < 8; ++i) {
            float o = __shfl_xor((float)mins[i], m, 32);
            mins[i] = fminf((float)mins[i], o);
        }
    }

    // VGPR i / half h holds row (i + 8*h): lanes rc<8 of each half emit rows
    if (rc < 8) {
        float v = mins[0];
        #pragma unroll
        for (int i = 1; i < 8; ++i) v = (rc == i) ? (float)mins[i] : v;
        rowmin[(size_t)b * Nrows + row_base + half * 8 + rc] = v;
    }
}

// out = 0.5 * ( mean(sqrt(d1)) + mean(sqrt(d2)) )
__global__ __launch_bounds__(512) void chamfer_finalize(
    const float* __restrict__ d1, const float* __restrict__ d2,
    float* __restrict__ out, int n1, int n2)
{
    __shared__ float s1s[512];
    __shared__ float s2s[512];
    float s1 = 0.0f, s2 = 0.0f;
    for (int i = threadIdx.x; i < n1; i += 512) s1 += sqrtf(d1[i]);
    for (int i = threadIdx.x; i < n2; i += 512) s2 += sqrtf(d2[i]);
    s1s[threadIdx.x] = s1;
    s2s[threadIdx.x] = s2;
    __syncthreads();
    for (int off = 256; off > 0; off >>= 1) {
        if ((int)threadIdx.x < off) {
            s1s[threadIdx.x] += s1s[threadIdx.x + off];
            s2s[threadIdx.x] += s2s[threadIdx.x + off];
        }
        __syncthreads();
    }
    if (threadIdx.x == 0)
        out[0] = 0.5f * (s1s[0] / (float)n1 + s2s[0] / (float)n2);
}

extern "C" void kernel_launch(void* const* d_in, const int* in_sizes, int n_in,
                              void* d_out, int out_size, void* d_ws, size_t ws_size,
                              hipStream_t stream)
{
    const float* y_pred = (const float*)d_in[0];  // [B, N, 3]
    const float* y_true = (const float*)d_in[1];  // [B, M, 3]
    const int B = 4;
    const int N = in_sizes[0] / (B * 3);
    const int M = in_sizes[1] / (B * 3);

    float* d1 = (float*)d_ws;            // [B, N]
    float* d2 = d1 + (size_t)B * N;      // [B, M]

    // pass 1: min over true points for each pred point
    {
        const int waves  = B * (N / 16);
        const int blocks = (waves * 32) / 256;
        chamfer_rowmin_wmma<<<blocks, 256, 0, stream>>>(y_pred, y_true, d1, N, M);
    }
    // pass 2: roles swapped — min over pred points for each true point
    {
        const int waves  = B * (M / 16);
        const int blocks = (waves * 32) / 256;
        chamfer_rowmin_wmma<<<blocks, 256, 0, stream>>>(y_true, y_pred, d2, M, N);
    }
    chamfer_finalize<<<1, 512, 0, stream>>>(d1, d2, (float*)d_out, B * N, B * M);
}